// GNNFailurePropagationModel_43207370998181
// MI455X (gfx1250) — compile-verified
//
#include <hip/hip_runtime.h>

#define NN 50000
#define EE 800000
#define ETOT (EE + NN)   // 850000 edges incl. self loops
#define HEADS 4
#define CH 64
#define NEG_SLOPE 0.2f

static_assert(NN % 16 == 0, "M dimension tiles exactly; GEMM has no tail");

typedef __attribute__((ext_vector_type(2))) float v2f;
typedef __attribute__((ext_vector_type(8))) float v8f;

// ---------------- helpers ----------------

__device__ __forceinline__ void edge_sd(const long long* __restrict__ ei, int e, int& s, int& d) {
  if (e < EE) { s = (int)ei[e]; d = (int)ei[EE + e]; }
  else        { s = e - EE; d = e - EE; }
}

// monotone float max via sign-split integer atomics
__device__ __forceinline__ void atomicMaxF(float* addr, float val) {
  int iv = __float_as_int(val);
  if (iv >= 0) atomicMax((int*)addr, iv);
  else         atomicMin((unsigned int*)addr, (unsigned int)iv);
}

__global__ void k_fill(float* __restrict__ p, float v, int n) {
  int i = blockIdx.x * blockDim.x + threadIdx.x;
  if (i < n) p[i] = v;
}

// ---------------- dense transform: O[NN,256] = A[NN,K] @ B[K,256] (f32 WMMA) ----------------
// block (32,8): 8 waves; wave wy computes output column tiles wy and wy+8.
// NN % 16 == 0 -> no row clamp, no store guard: straight-line epilogue.
template<int K>
__global__ void k_gemm_wmma(const float* __restrict__ A, const float* __restrict__ B,
                            float* __restrict__ O) {
  const int lane = threadIdx.x;            // 0..31
  const int wy   = threadIdx.y;            // 0..7
  const int mt   = blockIdx.x;             // M tile
  const int m    = lane & 15;
  const int kh   = (lane >> 4) << 1;       // 0 or 2 (K pair select)
  const int row  = mt * 16 + m;
  const int n    = lane & 15;
  const int c0   = wy * 16 + n;
  const int c1   = (wy + 8) * 16 + n;

  v8f acc0 = {0.f,0.f,0.f,0.f,0.f,0.f,0.f,0.f};
  v8f acc1 = {0.f,0.f,0.f,0.f,0.f,0.f,0.f,0.f};
  #pragma unroll
  for (int k0 = 0; k0 < K; k0 += 4) {
    v2f a, b0, b1;
    a[0]  = A[(long long)row * K + k0 + kh];
    a[1]  = A[(long long)row * K + k0 + kh + 1];
    b0[0] = B[(k0 + kh) * 256 + c0];
    b0[1] = B[(k0 + kh + 1) * 256 + c0];
    b1[0] = B[(k0 + kh) * 256 + c1];
    b1[1] = B[(k0 + kh + 1) * 256 + c1];
    acc0 = __builtin_amdgcn_wmma_f32_16x16x4_f32(false, a, false, b0, (short)0, acc0, false, false);
    acc1 = __builtin_amdgcn_wmma_f32_16x16x4_f32(false, a, false, b1, (short)0, acc1, false, false);
  }
  const int mo = (lane >> 4) << 3;         // C/D layout: lanes 16-31 hold M+8
  #pragma unroll
  for (int v = 0; v < 8; ++v) {
    const long long r = mt * 16 + mo + v;
    O[r * 256 + c0] = acc0[v];
    O[r * 256 + c1] = acc1[v];
  }
}

// ---------------- attention coefficients e_src/e_dst[n,h] ----------------
__global__ void k_attn_coef(const float* __restrict__ hfull,
                            const float* __restrict__ as, const float* __restrict__ ad,
                            float* __restrict__ esrc, float* __restrict__ edst) {
  int idx = blockIdx.x * blockDim.x + threadIdx.x;  // n*HEADS + h
  if (idx >= NN * HEADS) return;
  int n = idx / HEADS, h = idx % HEADS;
  const float* hp = hfull + ((long long)n * HEADS + h) * CH;
  float ss = 0.f, sd = 0.f;
  #pragma unroll 8
  for (int c = 0; c < CH; ++c) {
    float v = hp[c];
    ss += v * as[h * CH + c];
    sd += v * ad[h * CH + c];
  }
  esrc[idx] = ss; edst[idx] = sd;
}

// ---------------- edge logits (+leaky relu) & per-dst max ----------------
template<int H>
__global__ void k_edge_logits(const long long* __restrict__ ei,
                              const float* __restrict__ esrc, const float* __restrict__ edst,
                              float* __restrict__ logits, float* __restrict__ mmax) {
  int e = blockIdx.x * blockDim.x + threadIdx.x;
  if (e >= ETOT) return;
  int s, d; edge_sd(ei, e, s, d);
  #pragma unroll
  for (int h = 0; h < H; ++h) {
    float l = esrc[s * H + h] + edst[d * H + h];
    l = l > 0.f ? l : NEG_SLOPE * l;
    logits[(long long)e * H + h] = l;
    atomicMaxF(&mmax[d * H + h], l);
  }
}

// ---------------- exp(l - m) & per-dst denom ----------------
template<int H>
__global__ void k_edge_exp(const long long* __restrict__ ei,
                           const float* __restrict__ mmax,
                           float* __restrict__ exbuf, float* __restrict__ denom) {
  int e = blockIdx.x * blockDim.x + threadIdx.x;
  if (e >= ETOT) return;
  int s, d; edge_sd(ei, e, s, d);
  (void)s;
  #pragma unroll
  for (int h = 0; h < H; ++h) {
    float ex = expf(exbuf[(long long)e * H + h] - mmax[d * H + h]);
    exbuf[(long long)e * H + h] = ex;
    atomicAdd(&denom[d * H + h], ex);
  }
}

// ---------------- aggregation: one wave32 per (edge, head); lane moves float2 ----------------
template<int H>
__global__ void k_aggregate(const long long* __restrict__ ei,
                            const float* __restrict__ hfull,
                            const float* __restrict__ exbuf, const float* __restrict__ denom,
                            float* __restrict__ accum) {
  long long gtid = (long long)blockIdx.x * blockDim.x + threadIdx.x;
  long long wave = gtid >> 5;
  int lane = (int)(gtid & 31);
  if (wave >= (long long)ETOT * H) return;
  int e = (int)(wave / H);
  int h = (int)(wave % H);
  int s, d; edge_sd(ei, e, s, d);
  float alpha = exbuf[(long long)e * H + h] / (denom[d * H + h] + 1e-16f);
  const float2* hp = (const float2*)(hfull + ((long long)s * H + h) * CH);
  float2 v = hp[lane];
  float* ap = accum + ((long long)d * H + h) * CH + lane * 2;
  atomicAdd(ap,     alpha * v.x);
  atomicAdd(ap + 1, alpha * v.y);
}

// ---------------- finalize: mean over heads + bias, ELU ----------------
__global__ void k_finalize_elu(const float* __restrict__ accum, const float* __restrict__ b,
                               float* __restrict__ hnode) {
  int idx = blockIdx.x * blockDim.x + threadIdx.x;  // n*CH + c
  if (idx >= NN * CH) return;
  int n = idx / CH, c = idx % CH;
  float s = 0.f;
  #pragma unroll
  for (int h = 0; h < HEADS; ++h) s += accum[((long long)n * HEADS + h) * CH + c];
  s = s * (1.0f / HEADS) + b[c];
  hnode[idx] = s > 0.f ? s : (expf(s) - 1.0f);
}

// ---------------- layer 3 (H=1, C=1): linear + attn coefficients ----------------
__global__ void k_linear_attn3(const float* __restrict__ A, const float* __restrict__ W,
                               const float* __restrict__ as3, const float* __restrict__ ad3,
                               float* __restrict__ h3, float* __restrict__ es, float* __restrict__ ed) {
  int n = blockIdx.x * blockDim.x + threadIdx.x;
  if (n >= NN) return;
  float s = 0.f;
  #pragma unroll 8
  for (int k = 0; k < CH; ++k) s += A[(long long)n * CH + k] * W[k];
  h3[n] = s;
  es[n] = s * as3[0];
  ed[n] = s * ad3[0];
}

__global__ void k_agg_scalar(const long long* __restrict__ ei, const float* __restrict__ h3,
                             const float* __restrict__ exbuf, const float* __restrict__ denom,
                             float* __restrict__ accum) {
  int e = blockIdx.x * blockDim.x + threadIdx.x;
  if (e >= ETOT) return;
  int s, d; edge_sd(ei, e, s, d);
  float alpha = exbuf[e] / (denom[d] + 1e-16f);
  atomicAdd(&accum[d], alpha * h3[s]);
}

__global__ void k_finalize_sigmoid(const float* __restrict__ accum, const float* __restrict__ b,
                                   float* __restrict__ out) {
  int n = blockIdx.x * blockDim.x + threadIdx.x;
  if (n >= NN) return;
  float s = accum[n] + b[0];
  out[n] = 1.0f / (1.0f + expf(-s));
}

// ---------------- host launch ----------------

static inline int cdiv(long long a, long long b) { return (int)((a + b - 1) / b); }

extern "C" void kernel_launch(void* const* d_in, const int* in_sizes, int n_in,
                              void* d_out, int out_size, void* d_ws, size_t ws_size,
                              hipStream_t stream) {
  (void)in_sizes; (void)n_in; (void)out_size; (void)ws_size;
  const float*     x   = (const float*)d_in[0];
  const long long* ei  = (const long long*)d_in[1];   // int64 [2,E]
  const float* W1  = (const float*)d_in[2];
  const float* b1  = (const float*)d_in[3];
  const float* as1 = (const float*)d_in[4];
  const float* ad1 = (const float*)d_in[5];
  const float* W2  = (const float*)d_in[6];
  const float* b2  = (const float*)d_in[7];
  const float* as2 = (const float*)d_in[8];
  const float* ad2 = (const float*)d_in[9];
  const float* W3  = (const float*)d_in[10];
  const float* b3  = (const float*)d_in[11];
  const float* as3 = (const float*)d_in[12];
  const float* ad3 = (const float*)d_in[13];
  float* out = (float*)d_out;

  float* ws = (float*)d_ws;
  size_t off = 0;
  float* hfull = ws + off; off += (size_t)NN * HEADS * CH;   // [N,H,C] (layer3: h3 in [0,N))
  float* hnode = ws + off; off += (size_t)NN * CH;           // [N,C]
  float* esrc  = ws + off; off += (size_t)NN * HEADS;
  float* edst  = ws + off; off += (size_t)NN * HEADS;
  float* mmax  = ws + off; off += (size_t)NN * HEADS;
  float* denom = ws + off; off += (size_t)NN * HEADS;
  float* exbuf = ws + off; off += (size_t)ETOT * HEADS;
  float* accum = ws + off; off += (size_t)NN * HEADS * CH;

  const float NEG_INF = -3.402823466e38f;
  const dim3 gblk(32, 8);
  const int TB = 256;

  // ---------- Layer 1: in=8 -> [N,4,64] ----------
  k_fill<<<cdiv(NN*HEADS*CH, TB), TB, 0, stream>>>(accum, 0.f, NN*HEADS*CH);
  k_fill<<<cdiv(NN*HEADS,    TB), TB, 0, stream>>>(denom, 0.f, NN*HEADS);
  k_fill<<<cdiv(NN*HEADS,    TB), TB, 0, stream>>>(mmax,  NEG_INF, NN*HEADS);
  k_gemm_wmma<8><<<NN/16, gblk, 0, stream>>>(x, W1, hfull);
  k_attn_coef<<<cdiv(NN*HEADS, TB), TB, 0, stream>>>(hfull, as1, ad1, esrc, edst);
  k_edge_logits<HEADS><<<cdiv(ETOT, TB), TB, 0, stream>>>(ei, esrc, edst, exbuf, mmax);
  k_edge_exp<HEADS><<<cdiv(ETOT, TB), TB, 0, stream>>>(ei, mmax, exbuf, denom);
  k_aggregate<HEADS><<<cdiv((long long)ETOT*HEADS*32, TB), TB, 0, stream>>>(ei, hfull, exbuf, denom, accum);
  k_finalize_elu<<<cdiv(NN*CH, TB), TB, 0, stream>>>(accum, b1, hnode);

  // ---------- Layer 2: 64 -> [N,4,64] ----------
  k_fill<<<cdiv(NN*HEADS*CH, TB), TB, 0, stream>>>(accum, 0.f, NN*HEADS*CH);
  k_fill<<<cdiv(NN*HEADS,    TB), TB, 0, stream>>>(denom, 0.f, NN*HEADS);
  k_fill<<<cdiv(NN*HEADS,    TB), TB, 0, stream>>>(mmax,  NEG_INF, NN*HEADS);
  k_gemm_wmma<64><<<NN/16, gblk, 0, stream>>>(hnode, W2, hfull);
  k_attn_coef<<<cdiv(NN*HEADS, TB), TB, 0, stream>>>(hfull, as2, ad2, esrc, edst);
  k_edge_logits<HEADS><<<cdiv(ETOT, TB), TB, 0, stream>>>(ei, esrc, edst, exbuf, mmax);
  k_edge_exp<HEADS><<<cdiv(ETOT, TB), TB, 0, stream>>>(ei, mmax, exbuf, denom);
  k_aggregate<HEADS><<<cdiv((long long)ETOT*HEADS*32, TB), TB, 0, stream>>>(ei, hfull, exbuf, denom, accum);
  k_finalize_elu<<<cdiv(NN*CH, TB), TB, 0, stream>>>(accum, b2, hnode);

  // ---------- Layer 3: 64 -> 1, sigmoid ----------
  k_fill<<<cdiv(NN, TB), TB, 0, stream>>>(accum, 0.f, NN);
  k_fill<<<cdiv(NN, TB), TB, 0, stream>>>(denom, 0.f, NN);
  k_fill<<<cdiv(NN, TB), TB, 0, stream>>>(mmax,  NEG_INF, NN);
  k_linear_attn3<<<cdiv(NN, TB), TB, 0, stream>>>(hnode, W3, as3, ad3, hfull, esrc, edst);
  k_edge_logits<1><<<cdiv(ETOT, TB), TB, 0, stream>>>(ei, esrc, edst, exbuf, mmax);
  k_edge_exp<1><<<cdiv(ETOT, TB), TB, 0, stream>>>(ei, mmax, exbuf, denom);
  k_agg_scalar<<<cdiv(ETOT, TB), TB, 0, stream>>>(ei, hfull, exbuf, denom, accum);
  k_finalize_sigmoid<<<cdiv(NN, TB), TB, 0, stream>>>(accum, b3, out);
}